// ODDMoE_68101001445979
// MI455X (gfx1250) — compile-verified
//
#include <hip/hip_runtime.h>
#include <math.h>

#define HW 4096

typedef float v2f __attribute__((ext_vector_type(2)));
typedef float v8f __attribute__((ext_vector_type(8)));

static __device__ __forceinline__ v8f wmma_f32(v2f a, v2f b, v8f c) {
  // D = A(16x4 f32) * B(4x16 f32) + C(16x16 f32)
  return __builtin_amdgcn_wmma_f32_16x16x4_f32(false, a, false, b, (short)0, c, false, false);
}

static __device__ __forceinline__ float gelu_exact(float v) {
  return 0.5f * v * (1.0f + erff(v * 0.70710678118654752f));
}
static __device__ __forceinline__ float clamp_t(float v) {
  return fminf(fmaxf(v, 0.2f), 5.0f);
}

// ---------------------------------------------------------------------------
// Kernel A: pool + DCT band energies + coherence + router MLP -> gates,
// plus pre-collapsed FiLM combine coefficients gsum/bsum per (b,g).
// grid = 64 blocks (b*8+g), 256 threads.
// ---------------------------------------------------------------------------
__global__ __launch_bounds__(256) void kA_gates(
    const float* __restrict__ x, const float* __restrict__ bwh,
    const float* __restrict__ bwv, const float* __restrict__ bwd,
    const float* __restrict__ btemp, const float* __restrict__ rtemp,
    const float* __restrict__ wr1, const float* __restrict__ lnrg,
    const float* __restrict__ lnrb, const float* __restrict__ wr2,
    const float* __restrict__ br2, const float* __restrict__ fgam,
    const float* __restrict__ fbet,
    float* __restrict__ gate0, float* __restrict__ gsum, float* __restrict__ bsum)
{
  __shared__ float pool[256];
  int bg = blockIdx.x;
  int b = bg >> 3, g = bg & 7;
  int t = threadIdx.x;
  int u = t >> 4, v = t & 15;
  const float* xb = x + ((size_t)(b * 256 + g * 32)) * HW;
  float s = 0.f;
  for (int c = 0; c < 32; ++c) {
    const float* xc = xb + (size_t)c * HW + (u * 4) * 64 + v * 4;
    for (int i = 0; i < 4; ++i)
      for (int j = 0; j < 4; ++j) s += xc[i * 64 + j];
  }
  pool[t] = s * (1.0f / 512.0f);
  __syncthreads();

  if (t == 0) {
    const int PU[18] = {6,5,4,6,5,4, 0,0,0,1,1,1, 6,5,4,6,5,6};
    const int PV[18] = {0,0,0,1,1,1, 6,5,4,6,5,4, 6,5,4,5,6,4};
    const int MP[7]  = {0,2,0,0,10,12,15};  // python round(u*2.5), banker's
    const float PI = 3.14159265358979323846f;
    float resp[18];
    for (int k = 0; k < 18; ++k) {
      int ur = MP[PU[k]], vr = MP[PV[k]];
      float au = (ur == 0) ? 0.25f : 0.35355339059327373f;
      float av = (vr == 0) ? 0.25f : 0.35355339059327373f;
      float r = 0.f;
      for (int xx = 0; xx < 16; ++xx) {
        float cu = cosf(PI * (float)ur * (xx + 0.5f) / 16.0f);
        for (int yy = 0; yy < 16; ++yy) {
          float cv = cosf(PI * (float)vr * (yy + 0.5f) / 16.0f);
          r += pool[xx * 16 + yy] * cu * cv;
        }
      }
      resp[k] = au * av * r;
    }
    float Tb = clamp_t(btemp[0]);
    const float* bws[3] = {bwh, bwv, bwd};
    float E[3];
    for (int band = 0; band < 3; ++band) {
      float mx = -1e30f;
      for (int i = 0; i < 6; ++i) mx = fmaxf(mx, bws[band][i] / Tb);
      float se = 0.f, ex[6];
      for (int i = 0; i < 6; ++i) { ex[i] = expf(bws[band][i] / Tb - mx); se += ex[i]; }
      float acc = 0.f;
      for (int i = 0; i < 6; ++i) {
        float r = resp[band * 6 + i];
        acc += r * r * (ex[i] / se);
      }
      E[band] = acc;
    }
    float J11 = 0.f, J22 = 0.f, J12 = 0.f;
    for (int i = 0; i < 15; ++i)
      for (int j = 0; j < 15; ++j) {
        float gx = pool[i * 16 + j + 1] - pool[i * 16 + j];
        float gy = pool[(i + 1) * 16 + j] - pool[i * 16 + j];
        J11 += gx * gx; J22 += gy * gy; J12 += gx * gy;
      }
    J11 *= (1.f / 225.f); J22 *= (1.f / 225.f); J12 *= (1.f / 225.f);
    float disc = (J11 - J22) * (J11 - J22) + 4.0f * J12 * J12;
    float coh = sqrtf(disc + 1e-6f) / (J11 + J22 + 1e-6f);
    float mu = 0.f, sq = 0.f;
    for (int i = 0; i < 256; ++i) { mu += pool[i]; sq += pool[i] * pool[i]; }
    mu *= (1.f / 256.f); sq *= (1.f / 256.f);
    float sd = sqrtf(fmaxf(sq - mu * mu, 0.f));
    float Et = E[0] + E[1] + E[2];
    float feats[8] = {E[0], E[1], E[2], coh, mu, sd, Et, log1pf(Et)};
    float t1[16];
    for (int j = 0; j < 16; ++j) {
      float a = 0.f;
      for (int i = 0; i < 8; ++i) a += feats[i] * wr1[i * 16 + j];
      t1[j] = a;
    }
    float m16 = 0.f;
    for (int j = 0; j < 16; ++j) m16 += t1[j];
    m16 *= (1.f / 16.f);
    float v16 = 0.f;
    for (int j = 0; j < 16; ++j) { float d = t1[j] - m16; v16 += d * d; }
    v16 *= (1.f / 16.f);
    float iv16 = rsqrtf(v16 + 1e-5f);
    float hv[16];
    for (int j = 0; j < 16; ++j)
      hv[j] = gelu_exact((t1[j] - m16) * iv16 * lnrg[j] + lnrb[j]);
    float logits[5];
    for (int e = 0; e < 5; ++e) {
      float a = br2[e];
      for (int j = 0; j < 16; ++j) a += hv[j] * wr2[j * 5 + e];
      logits[e] = a;
    }
    float Tr = clamp_t(rtemp[0]);
    float mx = -1e30f;
    for (int e = 0; e < 5; ++e) mx = fmaxf(mx, logits[e] / Tr);
    float se = 0.f, ge[5];
    for (int e = 0; e < 5; ++e) { ge[e] = expf(logits[e] / Tr - mx); se += ge[e]; }
    for (int e = 0; e < 5; ++e) ge[e] /= se;
    gate0[bg] = ge[0];
    for (int c = 0; c < 32; ++c) {
      float gs = 0.f, bs = 0.f;
      for (int e = 0; e < 4; ++e) {
        gs += ge[1 + e] * fgam[e * 32 + c];
        bs += ge[1 + e] * fbet[e * 32 + c];
      }
      gsum[bg * 32 + c] = gs;
      bsum[bg * 32 + c] = bs;
    }
  }
}

// ---------------------------------------------------------------------------
// Kernel B: depthwise 3x3 conv statistics (per (bg, c) instance norm).
// grid = 8192 blocks (bg*32+c), 256 threads.
// ---------------------------------------------------------------------------
__global__ __launch_bounds__(256) void kB_dwstats(
    const float* __restrict__ x, const float* __restrict__ dwk,
    float* __restrict__ dwmean, float* __restrict__ dwvar)
{
  int blk = blockIdx.x;
  int c = blk & 31;
  const float* xc = x + ((size_t)blk) * HW;
  float kk[9];
  for (int i = 0; i < 9; ++i) kk[i] = dwk[c * 9 + i];
  int t = threadIdx.x;
  float s = 0.f, q = 0.f;
  for (int p = t; p < HW; p += 256) {
    int h = p >> 6, w = p & 63;
    float a = 0.f;
    for (int dy = 0; dy < 3; ++dy) {
      int hh = h + dy - 1;
      if (hh < 0 || hh >= 64) continue;
      for (int dx = 0; dx < 3; ++dx) {
        int wq = w + dx - 1;
        if (wq < 0 || wq >= 64) continue;
        a += kk[dy * 3 + dx] * xc[hh * 64 + wq];
      }
    }
    s += a; q += a * a;
  }
  __shared__ float rs[256], rq[256];
  rs[t] = s; rq[t] = q;
  __syncthreads();
  for (int off = 128; off > 0; off >>= 1) {
    if (t < off) { rs[t] += rs[t + off]; rq[t] += rq[t + off]; }
    __syncthreads();
  }
  if (t == 0) {
    float m = rs[0] * (1.f / 4096.f);
    dwmean[blk] = m;
    dwvar[blk] = rq[0] * (1.f / 4096.f) - m * m;
  }
}

// ---------------------------------------------------------------------------
// Kernel C (WMMA version): per-pixel expert math via three 16x32 @ 32x32
// f32 WMMA GEMMs per wave, gated combine, channel shuffle, coalesced store.
// grid = 4096 blocks (bg*64 + row), 128 threads = 4 waves x 16 pixels.
// ---------------------------------------------------------------------------
__global__ __launch_bounds__(128) void kC_combine(
    const float* __restrict__ x, const float* __restrict__ dwk,
    const float* __restrict__ gncg, const float* __restrict__ gncb,
    const float* __restrict__ pw, const float* __restrict__ lnsg,
    const float* __restrict__ lnsb, const float* __restrict__ w1,
    const float* __restrict__ w2, const float* __restrict__ dwmean,
    const float* __restrict__ dwvar, const float* __restrict__ gate0,
    const float* __restrict__ gsum, const float* __restrict__ bsum,
    float* __restrict__ buf)
{
  __shared__ float xs[32 * 3 * 64];   // 32 ch x 3 rows (halo) x 64 cols
  __shared__ float wtile[3 * 1024];   // w1 | w2 | pw^T   (B[k][n] layouts)
  __shared__ float prm[256];          // gncg|gncb|lnsg|lnsb|dwm|dwv|gsum|bsum
  __shared__ float dks[288];          // depthwise kernel
  __shared__ float slotA[4 * 544];    // per-wave y0n tile -> comb tile (pad 33)
  __shared__ float slotB[4 * 544];    // per-wave s tile -> h1 tile (pad 33)

  int blk = blockIdx.x;
  int bg = blk >> 6, row = blk & 63;
  int b = bg >> 3, g = bg & 7;
  int t = threadIdx.x;
  const float* xb = x + ((size_t)(bg * 32)) * HW;

  for (int idx = t; idx < 32 * 3 * 64; idx += 128) {
    int c = idx / 192;
    int rem = idx - c * 192;
    int r3 = rem >> 6, cc = rem & 63;
    int gh = row - 1 + r3;
    xs[idx] = (gh >= 0 && gh < 64) ? xb[(size_t)c * HW + gh * 64 + cc] : 0.0f;
  }
  for (int idx = t; idx < 1024; idx += 128) {
    wtile[idx] = w1[idx];
    wtile[1024 + idx] = w2[idx];
    int c = idx >> 5, o = idx & 31;
    wtile[2048 + c * 32 + o] = pw[o * 32 + c];   // pw^T: B[k=c][n=o]
  }
  for (int idx = t; idx < 256; idx += 128) {
    int c = idx & 31;
    float v;
    if (idx < 32)       v = gncg[c];
    else if (idx < 64)  v = gncb[c];
    else if (idx < 96)  v = lnsg[c];
    else if (idx < 128) v = lnsb[c];
    else if (idx < 160) v = dwmean[bg * 32 + c];
    else if (idx < 192) v = dwvar[bg * 32 + c];
    else if (idx < 224) v = gsum[bg * 32 + c];
    else                v = bsum[bg * 32 + c];
    prm[idx] = v;
  }
  for (int idx = t; idx < 288; idx += 128) dks[idx] = dwk[idx];
  __syncthreads();

  int lane = t & 31, wid = t >> 5;
  int nn = lane & 15;
  int koff = (lane < 16) ? 0 : 2;
  int radd = (lane < 16) ? 0 : 8;
  int colbase = wid * 16;
  float* sA = slotA + wid * 544;
  float* sB = slotB + wid * 544;

  // step 1: dwconv + per-channel instance norm + relu -> sA[px][c]
  {
    int px = lane & 15;
    int col = colbase + px;
    int c0 = (lane >> 4) * 16;
#pragma unroll
    for (int cc = 0; cc < 16; ++cc) {
      int c = c0 + cc;
      float a = 0.f;
#pragma unroll
      for (int dy = 0; dy < 3; ++dy) {
#pragma unroll
        for (int dx = 0; dx < 3; ++dx) {
          int wq = col + dx - 1;
          float xv = (wq >= 0 && wq < 64) ? xs[c * 192 + dy * 64 + wq] : 0.f;
          a += dks[c * 9 + dy * 3 + dx] * xv;
        }
      }
      float iv = rsqrtf(prm[160 + c] + 1e-5f);
      float vn = (a - prm[128 + c]) * iv * prm[0 + c] + prm[32 + c];
      sA[px * 33 + c] = fmaxf(vn, 0.f);
    }
  }
  // step 2: per-pixel LN -> sB[px][c]
  if (lane < 16) {
    int col = colbase + lane;
    float mu = 0.f;
    for (int c = 0; c < 32; ++c) mu += xs[c * 192 + 64 + col];
    mu *= (1.f / 32.f);
    float var = 0.f;
    for (int c = 0; c < 32; ++c) {
      float d = xs[c * 192 + 64 + col] - mu;
      var += d * d;
    }
    var *= (1.f / 32.f);
    float ivs = rsqrtf(var + 1e-5f);
    for (int c = 0; c < 32; ++c)
      sB[lane * 33 + c] = (xs[c * 192 + 64 + col] - mu) * ivs * prm[64 + c] + prm[96 + c];
  }

  v8f zero8 = {0.f, 0.f, 0.f, 0.f, 0.f, 0.f, 0.f, 0.f};

  // GEMM1: h1 = gelu(s @ w_seq1)      (16x32 = 16x32 @ 32x32)
  v8f acc0 = zero8, acc1 = zero8;
#pragma unroll
  for (int ks = 0; ks < 8; ++ks) {
    int k = ks * 4 + koff;
    v2f a;  a.x = sB[nn * 33 + k];           a.y = sB[nn * 33 + k + 1];
    v2f b0; b0.x = wtile[k * 32 + nn];       b0.y = wtile[(k + 1) * 32 + nn];
    v2f b1; b1.x = wtile[k * 32 + 16 + nn];  b1.y = wtile[(k + 1) * 32 + 16 + nn];
    acc0 = wmma_f32(a, b0, acc0);
    acc1 = wmma_f32(a, b1, acc1);
  }
#pragma unroll
  for (int r = 0; r < 8; ++r) {
    sB[(r + radd) * 33 + nn]      = gelu_exact(acc0[r]);
    sB[(r + radd) * 33 + 16 + nn] = gelu_exact(acc1[r]);
  }

  // GEMM2: delta = h1 @ w_seq2
  v8f dacc0 = zero8, dacc1 = zero8;
#pragma unroll
  for (int ks = 0; ks < 8; ++ks) {
    int k = ks * 4 + koff;
    v2f a;  a.x = sB[nn * 33 + k];                  a.y = sB[nn * 33 + k + 1];
    v2f b0; b0.x = wtile[1024 + k * 32 + nn];       b0.y = wtile[1024 + (k + 1) * 32 + nn];
    v2f b1; b1.x = wtile[1024 + k * 32 + 16 + nn];  b1.y = wtile[1024 + (k + 1) * 32 + 16 + nn];
    dacc0 = wmma_f32(a, b0, dacc0);
    dacc1 = wmma_f32(a, b1, dacc1);
  }

  // GEMM3: y0 = y0n @ pw^T
  v8f yacc0 = zero8, yacc1 = zero8;
#pragma unroll
  for (int ks = 0; ks < 8; ++ks) {
    int k = ks * 4 + koff;
    v2f a;  a.x = sA[nn * 33 + k];                  a.y = sA[nn * 33 + k + 1];
    v2f b0; b0.x = wtile[2048 + k * 32 + nn];       b0.y = wtile[2048 + (k + 1) * 32 + nn];
    v2f b1; b1.x = wtile[2048 + k * 32 + 16 + nn];  b1.y = wtile[2048 + (k + 1) * 32 + 16 + nn];
    yacc0 = wmma_f32(a, b0, yacc0);
    yacc1 = wmma_f32(a, b1, yacc1);
  }

  // combine: comb = gate0*y0 + (x + delta)*gsum + bsum  -> sA[px][c]
  float g0v = gate0[bg];
#pragma unroll
  for (int r = 0; r < 8; ++r) {
    int px = r + radd;
    int col = colbase + px;
    {
      int o = nn;
      float xv = xs[o * 192 + 64 + col];
      sA[px * 33 + o] = g0v * yacc0[r] + (xv + dacc0[r]) * prm[192 + o] + prm[224 + o];
    }
    {
      int o = 16 + nn;
      float xv = xs[o * 192 + 64 + col];
      sA[px * 33 + o] = g0v * yacc1[r] + (xv + dacc1[r]) * prm[192 + o] + prm[224 + o];
    }
  }
  __syncthreads();

  // coalesced shuffled store: c' = (cg%64)*4 + cg/64
  for (int idx = t; idx < 2048; idx += 128) {
    int c = idx >> 6, cpx = idx & 63;
    float vvv = slotA[(cpx >> 4) * 544 + (cpx & 15) * 33 + c];
    int cg = g * 32 + c;
    int cp = ((cg & 63) << 2) | (cg >> 6);
    buf[((size_t)(b * 256 + cp)) * HW + row * 64 + cpx] = vvv;
  }
}

// ---------------------------------------------------------------------------
// Kernel Z: zero gn_fuse stat accumulators.
// ---------------------------------------------------------------------------
__global__ void kZ_init(float* __restrict__ fsum, float* __restrict__ fsq)
{
  int t = threadIdx.x;
  if (t < 256) { fsum[t] = 0.f; fsq[t] = 0.f; }
}

// ---------------------------------------------------------------------------
// Kernel D: group-wise 64x64 GEMM (w_group_pw) via f32 WMMA, in-place on buf,
// plus gn_fuse partial statistics. grid = 1024 ((b*4+g4)*32 + pc), 256 thr.
// ---------------------------------------------------------------------------
__global__ __launch_bounds__(256) void kD_grouppw(
    const float* __restrict__ wgpw, float* __restrict__ buf,
    float* __restrict__ fsum, float* __restrict__ fsq)
{
  __shared__ float zt[64 * 128];   // K x N activations
  __shared__ float wt[64 * 64];    // M x K weights
  __shared__ float psum[8], psq[8];
  int blk = blockIdx.x;
  int pc = blk & 31, bg4 = blk >> 5;
  int b = bg4 >> 2, g4 = bg4 & 3;
  int p0 = pc * 128;
  int t = threadIdx.x;
  float* base = buf + ((size_t)(b * 256 + g4 * 64)) * HW;
  for (int idx = t; idx < 8192; idx += 256) {
    int ci = idx >> 7, pj = idx & 127;
    zt[idx] = base[(size_t)ci * HW + p0 + pj];
  }
  for (int idx = t; idx < 4096; idx += 256) wt[idx] = wgpw[g4 * 4096 + idx];
  if (t < 8) { psum[t] = 0.f; psq[t] = 0.f; }
  __syncthreads();

  int lane = t & 31, wid = t >> 5;
  int nt = wid * 16;
  int mrow = lane & 15;
  int koff = (lane < 16) ? 0 : 2;
  int radd = (lane < 16) ? 0 : 8;
  for (int mt = 0; mt < 4; ++mt) {
    v8f acc = {0.f, 0.f, 0.f, 0.f, 0.f, 0.f, 0.f, 0.f};
#pragma unroll
    for (int ks = 0; ks < 16; ++ks) {
      int k = ks * 4 + koff;
      v2f a, bb;
      a.x = wt[(mt * 16 + mrow) * 64 + k];
      a.y = wt[(mt * 16 + mrow) * 64 + k + 1];
      bb.x = zt[k * 128 + nt + mrow];
      bb.y = zt[(k + 1) * 128 + nt + mrow];
      acc = wmma_f32(a, bb, acc);
    }
#pragma unroll
    for (int r = 0; r < 8; ++r) {
      int orow = mt * 16 + r + radd;
      float vv = acc[r];
      base[(size_t)orow * HW + p0 + nt + mrow] = vv;
      atomicAdd(&psum[orow >> 3], vv);
      atomicAdd(&psq[orow >> 3], vv * vv);
    }
  }
  __syncthreads();
  if (t < 8) {
    atomicAdd(&fsum[b * 32 + g4 * 8 + t], psum[t]);
    atomicAdd(&fsq[b * 32 + g4 * 8 + t], psq[t]);
  }
}

// ---------------------------------------------------------------------------
// Kernel E: gn_fuse + relu fused into staging, 256x256 GEMM (w_fuse) via
// f32 WMMA with K-chunking, residual add. grid = 1024 ((b*2+mh)*64+pc).
// ---------------------------------------------------------------------------
__global__ __launch_bounds__(256) void kE_fuse(
    const float* __restrict__ x, const float* __restrict__ wfuse,
    const float* __restrict__ gnfg, const float* __restrict__ gnfb,
    const float* __restrict__ fsum, const float* __restrict__ fsq,
    const float* __restrict__ buf, float* __restrict__ out)
{
  __shared__ float ut[64 * 64];     // K-chunk x N activations (normed+relu)
  __shared__ float wt2[128 * 64];   // M x K-chunk weights
  __shared__ float gmean[32], grstd[32];
  int blk = blockIdx.x;
  int pc = blk & 63, bm = blk >> 6;
  int b = bm >> 1, mh = bm & 1;
  int p0 = pc * 64, mbase = mh * 128;
  int t = threadIdx.x;
  if (t < 32) {
    float m = fsum[b * 32 + t] * (1.f / 32768.f);
    gmean[t] = m;
    grstd[t] = rsqrtf(fsq[b * 32 + t] * (1.f / 32768.f) - m * m + 1e-5f);
  }
  int lane = t & 31, wid = t >> 5;
  int mrow = lane & 15;
  int koff = (lane < 16) ? 0 : 2;
  int radd = (lane < 16) ? 0 : 8;
  v8f zero = {0.f, 0.f, 0.f, 0.f, 0.f, 0.f, 0.f, 0.f};
  v8f acc[4] = {zero, zero, zero, zero};
  const float* ub = buf + (size_t)b * 256 * HW;
  for (int kc = 0; kc < 4; ++kc) {
    __syncthreads();
    for (int idx = t; idx < 4096; idx += 256) {
      int ci = idx >> 6, pj = idx & 63;
      int c = kc * 64 + ci;
      float v = ub[(size_t)c * HW + p0 + pj];
      int gg = c >> 3;
      v = (v - gmean[gg]) * grstd[gg] * gnfg[c] + gnfb[c];
      ut[idx] = fmaxf(v, 0.f);
    }
    for (int idx = t; idx < 8192; idx += 256) {
      int m = idx >> 6, ck = idx & 63;
      wt2[idx] = wfuse[(size_t)(mbase + m) * 256 + kc * 64 + ck];
    }
    if (kc < 3) {
      // warm L2/L0 for the next K-chunk (global_prefetch_b8)
      __builtin_prefetch((const void*)&ub[(size_t)((kc + 1) * 64 + (t >> 6)) * HW + p0 + (t & 63)], 0, 1);
    }
    __syncthreads();
#pragma unroll 4
    for (int ks = 0; ks < 16; ++ks) {
      int k = ks * 4 + koff;
      v2f a;
      a.x = wt2[(wid * 16 + mrow) * 64 + k];
      a.y = wt2[(wid * 16 + mrow) * 64 + k + 1];
#pragma unroll
      for (int ntv = 0; ntv < 4; ++ntv) {
        v2f bb;
        bb.x = ut[k * 64 + ntv * 16 + mrow];
        bb.y = ut[(k + 1) * 64 + ntv * 16 + mrow];
        acc[ntv] = wmma_f32(a, bb, acc[ntv]);
      }
    }
  }
#pragma unroll
  for (int ntv = 0; ntv < 4; ++ntv) {
#pragma unroll
    for (int r = 0; r < 8; ++r) {
      int o = mbase + wid * 16 + r + radd;
      size_t idx = ((size_t)(b * 256 + o)) * HW + p0 + ntv * 16 + mrow;
      out[idx] = x[idx] + acc[ntv][r];
    }
  }
}

// ---------------------------------------------------------------------------
extern "C" void kernel_launch(void* const* d_in, const int* in_sizes, int n_in,
                              void* d_out, int out_size, void* d_ws, size_t ws_size,
                              hipStream_t stream) {
  (void)in_sizes; (void)n_in; (void)out_size; (void)ws_size;
  const float* x      = (const float*)d_in[0];
  const float* bwh    = (const float*)d_in[1];
  const float* bwv    = (const float*)d_in[2];
  const float* bwd    = (const float*)d_in[3];
  const float* btemp  = (const float*)d_in[4];
  const float* rtemp  = (const float*)d_in[5];
  const float* lnsg   = (const float*)d_in[6];
  const float* lnsb   = (const float*)d_in[7];
  const float* w1     = (const float*)d_in[8];
  const float* w2     = (const float*)d_in[9];
  const float* fgam   = (const float*)d_in[10];
  const float* fbet   = (const float*)d_in[11];
  const float* dwk    = (const float*)d_in[12];
  const float* gncg   = (const float*)d_in[13];
  const float* gncb   = (const float*)d_in[14];
  const float* pw     = (const float*)d_in[15];
  const float* wr1    = (const float*)d_in[16];
  const float* lnrg   = (const float*)d_in[17];
  const float* lnrb   = (const float*)d_in[18];
  const float* wr2    = (const float*)d_in[19];
  const float* br2    = (const float*)d_in[20];
  const float* wgpw   = (const float*)d_in[21];
  const float* gnfg   = (const float*)d_in[22];
  const float* gnfb   = (const float*)d_in[23];
  const float* wfuse  = (const float*)d_in[24];
  float* out = (float*)d_out;

  float* ws     = (float*)d_ws;
  float* gate0  = ws;            // 64
  float* gsum   = ws + 64;       // 2048
  float* bsum   = ws + 2112;     // 2048
  float* dwmean = ws + 4160;     // 8192
  float* dwvar  = ws + 12352;    // 8192
  float* fsum   = ws + 20544;    // 256
  float* fsq    = ws + 20800;    // 256
  float* buf    = ws + 21056;    // 8*256*4096 floats

  kA_gates<<<64, 256, 0, stream>>>(x, bwh, bwv, bwd, btemp, rtemp, wr1, lnrg,
                                   lnrb, wr2, br2, fgam, fbet, gate0, gsum, bsum);
  kB_dwstats<<<8192, 256, 0, stream>>>(x, dwk, dwmean, dwvar);
  kC_combine<<<4096, 128, 0, stream>>>(x, dwk, gncg, gncb, pw, lnsg, lnsb, w1,
                                       w2, dwmean, dwvar, gate0, gsum, bsum, buf);
  kZ_init<<<1, 256, 0, stream>>>(fsum, fsq);
  kD_grouppw<<<1024, 256, 0, stream>>>(wgpw, buf, fsum, fsq);
  kE_fuse<<<1024, 256, 0, stream>>>(x, wfuse, gnfg, gnfb, fsum, fsq, buf, out);
}